// AutoRegressiveDistribution_68109591380876
// MI455X (gfx1250) — compile-verified
//
#include <hip/hip_runtime.h>
#include <math.h>

// Problem sizes (fixed by the reference)
#define BSZ 8192
#define DD  64
#define CTXN 256
#define HH  512

typedef __attribute__((ext_vector_type(2))) float v2f;
typedef __attribute__((ext_vector_type(8))) float v8f;

// ---------------------------------------------------------------------------
// Degree-sorted permutation of hidden units.
// mh[h] = h % 63 + 1. Units of degree m are h = (m-1) + 63*j,
// j = 0..8 for m<=8 (9 units), j = 0..7 for m>=9 (8 units).
// perm_h(pos): sorted position -> original hidden index.
// cumdeg(i): number of hidden units with degree <= i.
// ---------------------------------------------------------------------------
__device__ __forceinline__ int perm_h(int pos) {
  int m, j;
  if (pos < 72) { m = pos / 9 + 1; j = pos - (m - 1) * 9; }
  else          { int q = pos - 72; m = q / 8 + 9; j = q - (m - 9) * 8; }
  return (m - 1) + 63 * j;
}
__device__ __forceinline__ int cumdeg(int i) {
  return (i <= 8) ? 9 * i : 72 + 8 * (i - 8);
}

__device__ __forceinline__ float softplus_f(float x) {
  // jax.nn.softplus: max(x,0) + log1p(exp(-|x|))
  return fmaxf(x, 0.0f) + log1pf(expf(-fabsf(x)));
}

// Workspace layout (float offsets)
#define OFF_WCP 0         // [512][256]  Wc rows permuted
#define OFF_W1P 131072    // [512][64]   W1 rows permuted
#define OFF_W2P 163840    // [512][512]  W2 rows+cols permuted
#define OFF_WOP 425984    // [128][512]  Wo cols permuted
#define OFF_B1P 491520    // [512]
#define OFF_B2P 492032    // [512]
#define WS_FLOATS 492544

// ---------------------------------------------------------------------------
// Prep kernel: build degree-permuted weight copies. Masks M1/M2/Mo are
// implemented exactly by the degree-prefix summation ranges in the main
// kernel, so no explicit masking is needed here.
// ---------------------------------------------------------------------------
__global__ void made_prep_kernel(const float* __restrict__ W1,
                                 const float* __restrict__ b1,
                                 const float* __restrict__ Wc,
                                 const float* __restrict__ W2,
                                 const float* __restrict__ b2,
                                 const float* __restrict__ Wo,
                                 float* __restrict__ ws) {
  int idx = blockIdx.x * blockDim.x + threadIdx.x;
  if (idx < 131072) {                       // Wcp
    int pos = idx >> 8, c = idx & 255;
    ws[OFF_WCP + idx] = Wc[perm_h(pos) * CTXN + c];
  } else if (idx < 163840) {                // W1p
    int t = idx - 131072; int pos = t >> 6, d = t & 63;
    ws[OFF_W1P + t] = W1[perm_h(pos) * DD + d];
  } else if (idx < 425984) {                // W2p
    int t = idx - 163840; int pp = t >> 9, k = t & 511;
    ws[OFF_W2P + t] = W2[perm_h(pp) * HH + perm_h(k)];
  } else if (idx < 491520) {                // Wop
    int t = idx - 425984; int o = t >> 9, k = t & 511;
    ws[OFF_WOP + t] = Wo[o * HH + perm_h(k)];
  } else if (idx < 492032) {                // b1p
    int t = idx - 491520;
    ws[OFF_B1P + t] = b1[perm_h(t)];
  } else if (idx < WS_FLOATS) {             // b2p
    int t = idx - 492032;
    ws[OFF_B2P + t] = b2[perm_h(t)];
  }
}

// ---------------------------------------------------------------------------
// Main fused kernel. One workgroup = 64 batch rows, 256 threads (8 wave32).
// LDS (floats, padded stride 65 to avoid bank conflicts on [unit][row]):
//   h1buf [512][65]  = 33280   (holds ctx_h+b1, overwritten by relu'd h1)
//   h2buf [512][65]  = 33280   (context tile staged here first, stride 257)
//   zbuf  [64][64]   = 4096    z[d*64+b]
//   mubuf [64][64]   = 4096    mu[b*64+i]
//   scbuf [64][64]   = 4096    scale[b*64+i]
//   b2s   [512]      = 512
//   outtmp[4][64]    = 256     phase-3 partial dots
// total 79616 floats = 318464 bytes  (fits in 320KB WGP LDS)
// ---------------------------------------------------------------------------
#define LDS_H1 0
#define LDS_H2 33280
#define LDS_Z  66560
#define LDS_MU 70656
#define LDS_SC 74752
#define LDS_B2 78848
#define LDS_OT 79360
#define LDS_FLOATS 79616
#define LDS_BYTES (LDS_FLOATS * 4)

__launch_bounds__(256)
__global__ void made_rsample_kernel(const float* __restrict__ context,
                                    const float* __restrict__ eps,
                                    const float* __restrict__ bo,
                                    const float* __restrict__ ws,
                                    float* __restrict__ out) {
  extern __shared__ float smem[];
  float* h1buf  = smem + LDS_H1;
  float* h2buf  = smem + LDS_H2;
  float* zbuf   = smem + LDS_Z;
  float* mubuf  = smem + LDS_MU;
  float* scbuf  = smem + LDS_SC;
  float* b2s    = smem + LDS_B2;
  float* outtmp = smem + LDS_OT;

  const float* Wcp = ws + OFF_WCP;
  const float* W1p = ws + OFF_W1P;
  const float* W2p = ws + OFF_W2P;
  const float* Wop = ws + OFF_WOP;
  const float* b1p = ws + OFF_B1P;
  const float* b2p = ws + OFF_B2P;

  const int tid  = threadIdx.x;
  const int wave = tid >> 5;
  const int lane = tid & 31;
  const int b0 = blockIdx.x * 64;

  // ---- Phase A: stage 64x256 context tile into LDS (stride 257), load b2 ----
  float* ctxs = h2buf;  // reuse h2 region; 64*257 = 16448 <= 33280
  for (int idx = tid; idx < 64 * CTXN; idx += 256) {
    int r = idx >> 8, c = idx & 255;
    ctxs[r * 257 + c] = context[(size_t)(b0 + r) * CTXN + c];
  }
  for (int k = tid; k < HH; k += 256) b2s[k] = b2p[k];
  __syncthreads();

  // ---- Phase B: ctx_h (degree-permuted columns) via V_WMMA_F32_16X16X4_F32 ----
  // 128 tiles of 16x16 (4 row-tiles x 32 col-tiles), 16 tiles per wave.
  {
    const int M = lane & 15;               // output col (N) within tile / A row / B col
    const int koff = (lane < 16) ? 0 : 2;  // A/B VGPR0 holds K+0 | K+2
    for (int t = wave; t < 128; t += 8) {
      const int row0 = (t & 3) * 16;
      const int col0 = (t >> 2) * 16;
      v8f acc = {};
      for (int k0 = 0; k0 < CTXN; k0 += 4) {
        v2f a, b;
        const float* arow = &ctxs[(row0 + M) * 257 + k0 + koff];
        a[0] = arow[0];
        a[1] = arow[1];
        const float* brow = &Wcp[(col0 + M) * CTXN + k0 + koff];  // B[k][n] = Wcp[n][k]
        b[0] = brow[0];
        b[1] = brow[1];
        acc = __builtin_amdgcn_wmma_f32_16x16x4_f32(
            false, a, false, b, (short)0, acc, false, false);
      }
      // C/D layout: VGPR v, lane L: M = v + (L<16?0:8), N = L&15
      const int mbase = row0 + ((lane < 16) ? 0 : 8);
      const float bias = b1p[col0 + M];
#pragma unroll
      for (int v = 0; v < 8; ++v) {
        h1buf[(col0 + M) * 65 + mbase + v] = acc[v] + bias;
      }
    }
  }
  __syncthreads();

  // ---- Step 0: outputs of degree 1 depend on no hidden units ----
  if (tid < 64) {
    int b = tid;
    float mu0 = bo[0];
    float sc0 = softplus_f(bo[DD + 0]);
    mubuf[b * 64 + 0] = mu0;
    scbuf[b * 64 + 0] = sc0;
    zbuf[0 * 64 + b] = mu0 + sc0 * eps[(size_t)(b0 + b) * DD + 0];
  }
  __syncthreads();

  // ---- Sequential degree loop: each h1/h2 column computed exactly once ----
  for (int i = 1; i < DD; ++i) {
    const int c1 = (i <= 8) ? 9 : 8;     // #units of degree i
    const int p0 = cumdeg(i - 1);        // first sorted position of degree i
    const int P  = cumdeg(i);            // prefix length usable by outputs i

    // Phase 1 (VALU, tiny): h1 new columns: relu(ctx+b1 + sum_{d<i} z*W1p)
    for (int idx = tid; idx < c1 * 64; idx += 256) {
      int b = idx & 63, c = idx >> 6, pos = p0 + c;
      float acc = h1buf[pos * 65 + b];
      const float* w1r = W1p + pos * DD;
      for (int d = 0; d < i; ++d) acc += zbuf[d * 64 + b] * w1r[d];
      h1buf[pos * 65 + b] = fmaxf(acc, 0.0f);
    }
    __syncthreads();

    // Phase 2 (WMMA): h2[p0..P) = relu(b2 + h1[:,0:P] @ W2p[p0..P,0:P]^T)
    // Waves 0-3 each own one 16-row batch tile; single 16-col tile (c1<=9 valid).
    // A[M][k] = h1buf[k][row0+M] (LDS); B[k][n] = W2p[p0+n][k] (global, L2-hot),
    // zero-masked for k >= P so round-up tail / future columns contribute 0.
    if (wave < 4) {
      const int row0 = wave * 16;
      const int M = lane & 15;
      const int koff = (lane < 16) ? 0 : 2;
      const int rowN = min(p0 + M, HH - 1);    // clamp garbage B columns
      const float* w2c = W2p + rowN * HH;
      v8f acc = {};
      const int nK = (P + 3) & ~3;
      for (int k0 = 0; k0 < nK; k0 += 4) {
        const int k = k0 + koff;
        v2f a, b;
        a[0] = h1buf[k * 65 + row0 + M];
        a[1] = h1buf[(k + 1) * 65 + row0 + M];
        b[0] = (k < P) ? w2c[k] : 0.0f;
        b[1] = (k + 1 < P) ? w2c[k + 1] : 0.0f;
        acc = __builtin_amdgcn_wmma_f32_16x16x4_f32(
            false, a, false, b, (short)0, acc, false, false);
      }
      if (M < c1) {
        const float bias = b2s[p0 + M];
        const int mb = row0 + ((lane < 16) ? 0 : 8);
#pragma unroll
        for (int v = 0; v < 8; ++v) {
          h2buf[(p0 + M) * 65 + mb + v] = fmaxf(acc[v] + bias, 0.0f);
        }
      }
    }
    __syncthreads();

    // Phase 3a (VALU, all 256 threads): partial dots for mu / pre-scale,
    // 2-way split over K.
    {
      int b = tid & 63;
      int sel = (tid >> 6) & 1;
      int half = tid >> 7;
      int kb = half ? (P >> 1) : 0;
      int ke = half ? P : (P >> 1);
      int o = sel ? (DD + i) : i;
      const float* wor = Wop + o * HH;
      float acc = 0.0f;
      for (int k = kb; k < ke; ++k) acc += h2buf[k * 65 + b] * wor[k];
      outtmp[(half * 2 + sel) * 64 + b] = acc;
    }
    __syncthreads();

    // Phase 3b: combine partials, softplus, z-update (one thread per row)
    if (tid < 64) {
      int b = tid;
      float mu = bo[i]      + outtmp[0 * 64 + b] + outtmp[2 * 64 + b];
      float ps = bo[DD + i] + outtmp[1 * 64 + b] + outtmp[3 * 64 + b];
      float sc = softplus_f(ps);
      mubuf[b * 64 + i] = mu;
      scbuf[b * 64 + i] = sc;
      zbuf[i * 64 + b] = mu + sc * eps[(size_t)(b0 + b) * DD + i];
    }
    __syncthreads();
  }

  // ---- Final coalesced writes: z, mus.T, scales.T ----
  float* out_z  = out;
  float* out_mu = out + (size_t)BSZ * DD;
  float* out_sc = out + (size_t)2 * BSZ * DD;
  for (int idx = tid; idx < 64 * 64; idx += 256) {
    int b = idx >> 6, d = idx & 63;
    size_t g = (size_t)(b0 + b) * DD + d;
    out_z[g]  = zbuf[d * 64 + b];
    out_mu[g] = mubuf[b * 64 + d];
    out_sc[g] = scbuf[b * 64 + d];
  }
}

// ---------------------------------------------------------------------------
extern "C" void kernel_launch(void* const* d_in, const int* in_sizes, int n_in,
                              void* d_out, int out_size, void* d_ws, size_t ws_size,
                              hipStream_t stream) {
  const float* context = (const float*)d_in[0];
  const float* eps     = (const float*)d_in[1];
  const float* W1      = (const float*)d_in[2];
  const float* b1      = (const float*)d_in[3];
  const float* Wc      = (const float*)d_in[4];
  const float* W2      = (const float*)d_in[5];
  const float* b2      = (const float*)d_in[6];
  const float* Wo      = (const float*)d_in[7];
  const float* bo      = (const float*)d_in[8];
  float* out = (float*)d_out;
  float* ws  = (float*)d_ws;

  (void)in_sizes; (void)n_in; (void)out_size; (void)ws_size;

  // Allow 318464 B of dynamic LDS (WGP supports 320KB).
  (void)hipFuncSetAttribute((const void*)made_rsample_kernel,
                            hipFuncAttributeMaxDynamicSharedMemorySize,
                            LDS_BYTES);

  made_prep_kernel<<<WS_FLOATS / 256, 256, 0, stream>>>(W1, b1, Wc, W2, b2, Wo, ws);
  made_rsample_kernel<<<BSZ / 64, 256, LDS_BYTES, stream>>>(context, eps, bo, ws, out);
}